// MLHLoss2_73538430042375
// MI455X (gfx1250) — compile-verified
//
#include <hip/hip_runtime.h>
#include <hip/hip_bf16.h>
#include <math.h>

// CDNA5 / gfx1250: wave32. WMMA f32 16x16x4 operands:
//   A 16x4 f32 = 2 VGPRs/lane, B 4x16 f32 = 2 VGPRs/lane, C/D 16x16 f32 = 8 VGPRs/lane.
typedef float v2f __attribute__((ext_vector_type(2)));
typedef float v8f __attribute__((ext_vector_type(8)));

// Reference rho == 1.0 -> per-element weights are identically 1.0, so the
// weighted sum of squares equals the plain sum of squares. (The avg_grad /
// target_diff logic cannot change the result when RHO == 1.)
#define RHO 1.0f

__device__ __forceinline__ float wave_reduce_add(float v) {
#pragma unroll
    for (int off = 16; off > 0; off >>= 1)
        v += __shfl_xor(v, off, 32);
    return v;
}

__device__ __forceinline__ float wave_reduce_max(float v) {
#pragma unroll
    for (int off = 16; off > 0; off >>= 1)
        v = fmaxf(v, __shfl_xor(v, off, 32));
    return v;
}

// Stage 1: streaming pass over output/target. HBM-bound: b128 loads,
// grid-stride, per-lane register accumulation. 128 threads = 4 wave32s per
// block so the 4 wave partials map exactly onto the K=4 dimension of
// V_WMMA_F32_16X16X4_F32 for the block-level combine.
__global__ __launch_bounds__(128) void mlh_partial_kernel(
    const float* __restrict__ outp, const float* __restrict__ tgtp,
    float* __restrict__ pAbsSq,   // [2*gridDim.x] interleaved {abs,sq}
    float* __restrict__ pMax,     // [gridDim.x]
    long long n4, long long n) {
    const int tid = threadIdx.x;
    const long long gtid = (long long)blockIdx.x * blockDim.x + tid;
    const long long stride = (long long)gridDim.x * blockDim.x;

    float sAbs = 0.0f, sSq = 0.0f, vMax = -INFINITY;

    const float4* o4 = (const float4*)outp;
    const float4* t4 = (const float4*)tgtp;
    for (long long i = gtid; i < n4; i += stride) {
        float4 o = o4[i];
        float4 t = t4[i];
        float d0 = o.x - t.x, d1 = o.y - t.y, d2 = o.z - t.z, d3 = o.w - t.w;
        sAbs += fabsf(d0) + fabsf(d1) + fabsf(d2) + fabsf(d3);
        sSq  += d0 * d0 + d1 * d1 + d2 * d2 + d3 * d3;  // weights == 1 (RHO==1)
        vMax = fmaxf(vMax, fmaxf(fmaxf(t.x, t.y), fmaxf(t.z, t.w)));
    }
    // scalar tail (n not multiple of 4)
    for (long long i = 4 * n4 + gtid; i < n; i += stride) {
        float d = outp[i] - tgtp[i];
        sAbs += fabsf(d);
        sSq  += d * d;
        vMax = fmaxf(vMax, tgtp[i]);
    }

    // lane reduce within each wave32
    sAbs = wave_reduce_add(sAbs);
    sSq  = wave_reduce_add(sSq);
    vMax = wave_reduce_max(vMax);

    __shared__ float Lsum[8];  // [2w]=abs of wave w, [2w+1]=sq of wave w
    __shared__ float Lmax[4];
    const int wave = tid >> 5;
    const int lane = tid & 31;
    if (lane == 0) {
        Lsum[2 * wave]     = sAbs;
        Lsum[2 * wave + 1] = sSq;
        Lmax[wave]         = vMax;
    }
    __syncthreads();

    if (tid < 32) {  // wave 0: EXEC is all 1s across its 32 lanes (WMMA requirement)
        // Ones-matmul combine: D = ONES(16x4) x B(4x16) + 0 => D[m][n] = sum_k B[k][n].
        // Pack B[k][0] = abs partial of wave k, B[k][1] = sq partial of wave k.
        // B layout (4x16 f32, 2 VGPRs): b.x lanes 0-15 = row k=0, lanes 16-31 = k=1;
        //                               b.y lanes 0-15 = row k=2, lanes 16-31 = k=3.
        const int row = lane >> 4;   // 0 or 1
        const int col = lane & 15;   // N index
        v2f a; a.x = 1.0f; a.y = 1.0f;                 // A = ones(16x4)
        v2f b;
        b.x = (col < 2) ? Lsum[2 * row + col]       : 0.0f;
        b.y = (col < 2) ? Lsum[2 * (row + 2) + col] : 0.0f;
        v8f c = {};
        // 8 args: (neg_a, A, neg_b, B, c_mod, C, reuse_a, reuse_b)
        c = __builtin_amdgcn_wmma_f32_16x16x4_f32(
            false, a, false, b, (short)0, c, false, false);
        // D layout: lane n (0-15), VGPR 0 holds D[0][n] -> lane 0: block abs, lane 1: block sq.
        if (lane < 2)
            pAbsSq[2 * (long long)blockIdx.x + lane] = c[0];
        if (lane == 0)
            pMax[blockIdx.x] =
                fmaxf(fmaxf(Lmax[0], Lmax[1]), fmaxf(Lmax[2], Lmax[3]));
    }
}

// Stage 2: single wave32 reduces the block partials and writes the scalar.
__global__ __launch_bounds__(32) void mlh_final_kernel(
    const float* __restrict__ pAbsSq, const float* __restrict__ pMax,
    int nPart, float invN, float* __restrict__ result) {
    const int lane = threadIdx.x;
    float sAbs = 0.0f, sSq = 0.0f, vMax = -INFINITY;
    for (int i = lane; i < nPart; i += 32) {
        sAbs += pAbsSq[2 * i];
        sSq  += pAbsSq[2 * i + 1];
        vMax = fmaxf(vMax, pMax[i]);
    }
    sAbs = wave_reduce_add(sAbs);
    sSq  = wave_reduce_add(sSq);
    vMax = wave_reduce_max(vMax);
    if (lane == 0) {
        float mae  = sAbs * invN;
        float mse  = sSq;             // weighted_loss with unit weights
        float rmse = sqrtf(mse);
        result[0] = mae + mse + rmse + rmse / vMax;
    }
}

extern "C" void kernel_launch(void* const* d_in, const int* in_sizes, int n_in,
                              void* d_out, int out_size, void* d_ws, size_t ws_size,
                              hipStream_t stream) {
    // setup_inputs order: inputs (unused by reference), output, target
    const float* outp = (const float*)d_in[1];
    const float* tgtp = (const float*)d_in[2];
    const long long n  = (long long)in_sizes[1];
    const long long n4 = n / 4;

    const int blocks  = 1024;  // 1024 blocks x 4 waves = 4096 wave32s in flight
    const int threads = 128;

    float* pAbsSq = (float*)d_ws;           // 2*blocks floats
    float* pMax   = pAbsSq + 2 * blocks;    // blocks floats

    mlh_partial_kernel<<<blocks, threads, 0, stream>>>(outp, tgtp, pAbsSq, pMax,
                                                       n4, n);
    mlh_final_kernel<<<1, 32, 0, stream>>>(pAbsSq, pMax, blocks,
                                           1.0f / (float)n, (float*)d_out);
}